// DOSLoss_34660386079100
// MI455X (gfx1250) — compile-verified
//
#include <hip/hip_runtime.h>
#include <math.h>

typedef __attribute__((ext_vector_type(2))) float v2f;
typedef __attribute__((ext_vector_type(8))) float v8f;

#define N_DOS 400
#define ZLO 179            // ZERO_IDX - 20
#define ZHI 219            // ZERO_IDX + 20 (exclusive)
#define DXC (20.0f / 399.0f)
#define WAVES_PER_BLOCK 8
#define ROWS_PER_WAVE 16
#define BLOCK_THREADS 256
#define TAB_STRIDE 404     // mod 64 == 20 -> 4-bank groups per lane are disjoint

// Lane-half swap (xor 16) via ds_swizzle with immediate SWAPX16 pattern:
// offset = xor_mask(0x10)<<10 | or_mask(0)<<5 | and_mask(0x1f) = 0x401F
__device__ __forceinline__ float swz16(float v) {
    return __int_as_float(__builtin_amdgcn_ds_swizzle(__float_as_int(v), 0x401F));
}

// Main pass: one read of out/scaled, per-block partial sums of the 4 loss terms.
// Moments m0..m4 AND the EF-window sum (column 5) are computed by a single
// [16,400]x[400,16] WMMA chain against a constant Vandermonde+indicator table.
__global__ __launch_bounds__(BLOCK_THREADS)
void dos_loss_main(const float* __restrict__ outp,     // [B,400]
                   const float* __restrict__ scaling,  // [B]
                   const float* __restrict__ scaled,   // [B,400]
                   const float* __restrict__ sfac,     // [B]
                   const float* __restrict__ features, // [B,5]
                   float* __restrict__ partials)       // [gridDim.x][4]
{
    __shared__ float tabB[16 * TAB_STRIDE];              // [n][k]: x_k^n, n<5; window, n==5; else 0
    __shared__ float cTile[WAVES_PER_BLOCK][16 * 16];    // WMMA accumulator spill
    __shared__ float wred[WAVES_PER_BLOCK][4];

    const int tid = threadIdx.x;

    // Build transposed B table (row-contiguous in k -> one ds_load_b64 per iter).
    for (int i = tid; i < 16 * TAB_STRIDE; i += BLOCK_THREADS) {
        int nn = i / TAB_STRIDE, k = i - nn * TAB_STRIDE;
        float p = 0.0f;
        if (k < N_DOS) {
            float x = -10.0f + (float)k * DXC;
            if (nn < 5)       { p = 1.0f; for (int j = 0; j < nn; ++j) p *= x; }
            else if (nn == 5) { p = (k >= ZLO && k < ZHI) ? 1.0f : 0.0f; }
        }
        tabB[i] = p;
    }
    __syncthreads();

    const int wave = tid >> 5;
    const int lane = tid & 31;
    const int h    = lane >> 4;     // half-wave: column-pair select
    const int n    = lane & 15;     // row within tile / moment column
    const float hsel = h ? 1.0f : 0.0f;
    const long long rowBase = ((long long)blockIdx.x * WAVES_PER_BLOCK + wave) * ROWS_PER_WAVE;
    const long long row = rowBase + n;           // lanes l and l+16 co-own row l

    const float srow = scaling[row];
    const float* orow = outp   + row * N_DOS;
    const float* zrow = scaled + row * N_DOS;
    const float* tb   = tabB + n * TAB_STRIDE + 2 * h;   // lane's B-table base

    v8f acc = {0.f,0.f,0.f,0.f,0.f,0.f,0.f,0.f};
    float accDos = 0.f, accCum = 0.f, carry = 0.f;

    for (int k0 = 0; k0 < N_DOS; k0 += 4) {
        const int c0 = k0 + 2 * h;               // this lane's first column

        // Uniform-branch streaming prefetch (gfx1250 global_prefetch_b8):
        // 16 rows x 128B lines, 128 columns ahead. k0 is wave-uniform so EXEC
        // stays all-ones for the WMMAs below.
        if ((k0 & 31) == 0) {
            __builtin_prefetch(orow + c0 + 128, 0, 1);
            __builtin_prefetch(zrow + c0 + 128, 0, 1);
        }

        v2f o = *(const v2f*)(orow + c0);        // global_load_b64
        v2f z = *(const v2f*)(zrow + c0);

        // dos L1 term
        float d0 = o.x - z.x, d1 = o.y - z.y;
        accDos += fabsf(d0) + fabsf(d1);

        // cumsum L1 term: cumsum(out)-cumsum(scaled) == prefix(out-scaled).
        // Row l is split lane l (K0,K1) / lane l+16 (K2,K3): exchange pair sums.
        float pair  = d0 + d1;
        float other = swz16(pair);               // low gets s23, high gets s01
        float base  = fmaf(hsel, other, carry);  // high half adds s01
        float p0 = base + d0;
        float p1 = p0 + d1;
        accCum += fabsf(p0) + fabsf(p1);
        carry  += pair + other;                  // identical in both halves

        // dos = |out * scaling| feeds WMMA A operand (16x4 f32 tile)
        v2f A; A.x = fabsf(o.x * srow); A.y = fabsf(o.y * srow);

        // B[k][n]: one b64 load of two contiguous table entries for this lane
        v2f Bv = *(const v2f*)(tb + k0);

        // D[m][n] += sum_k dos[m,k]*x_k^n : cols 0..4 = raw moments, col 5 = EF sum
        acc = __builtin_amdgcn_wmma_f32_16x16x4_f32(
                  false, A, false, Bv, (short)0, acc, false, false);
    }

    // Spill accumulator tile: VGPR j = (M=j, lanes 0-15) / (M=j+8, lanes 16-31)
    float* ct = cTile[wave];
#pragma unroll
    for (int j = 0; j < 8; ++j)
        ct[(j + 8 * h) * 16 + n] = acc[j];

    __syncthreads();

    float featAbs = 0.f, scalAbs = 0.f;
    if (h == 0) {                                 // lane n handles row rowBase+n
        float m0 = ct[n * 16 + 0], m1 = ct[n * 16 + 1], m2 = ct[n * 16 + 2];
        float m3 = ct[n * 16 + 3], m4 = ct[n * 16 + 4], m5 = ct[n * 16 + 5];
        float inv = 1.0f / m0;
        float c  = m1 * inv;
        float c2 = c * c;
        float wid = m2 * inv - c2;
        float mu3 = (m3 - 3.f*c*m2 + 3.f*c2*m1 - c2*c*m0) * inv;
        float mu4 = (m4 - 4.f*c*m3 + 6.f*c2*m2 - 4.f*c2*c*m1 + c2*c2*m0) * inv;
        float skew = mu3 / (wid * sqrtf(wid));
        float kurt = mu4 / (wid * wid);
        float ef   = m5 * DXC;                    // EF window folded into GEMM col 5
        const float* fr = features + row * 5;
        featAbs = fabsf(fr[0] - c) + fabsf(fr[1] - wid) + fabsf(fr[2] - skew)
                + fabsf(fr[3] - kurt) + fabsf(fr[4] - ef);
        scalAbs = fabsf(srow - sfac[row]);
    }

    // Wave reduce 4 components, then per-block partials (deterministic, no atomics)
    float v0 = accDos, v1 = accCum, v2 = featAbs, v3 = scalAbs;
#pragma unroll
    for (int off = 16; off > 0; off >>= 1) {
        v0 += __shfl_xor(v0, off, 32);
        v1 += __shfl_xor(v1, off, 32);
        v2 += __shfl_xor(v2, off, 32);
        v3 += __shfl_xor(v3, off, 32);
    }
    if (lane == 0) { wred[wave][0]=v0; wred[wave][1]=v1; wred[wave][2]=v2; wred[wave][3]=v3; }
    __syncthreads();
    if (tid < 4) {
        float s = 0.f;
#pragma unroll
        for (int w = 0; w < WAVES_PER_BLOCK; ++w) s += wred[w][tid];
        partials[blockIdx.x * 4 + tid] = s;
    }
}

// Final deterministic combine of per-block partials.
__global__ __launch_bounds__(256)
void dos_loss_finish(const float* __restrict__ partials, int nblocks,
                     long long Brows, float* __restrict__ out_scalar)
{
    __shared__ float red[256];
    float a0 = 0.f, a1 = 0.f, a2 = 0.f, a3 = 0.f;
    for (int b = threadIdx.x; b < nblocks; b += 256) {
        a0 += partials[b * 4 + 0];
        a1 += partials[b * 4 + 1];
        a2 += partials[b * 4 + 2];
        a3 += partials[b * 4 + 3];
    }
    float comp[4] = {a0, a1, a2, a3};
    float res[4];
#pragma unroll
    for (int k = 0; k < 4; ++k) {
        red[threadIdx.x] = comp[k];
        __syncthreads();
        for (int s = 128; s > 0; s >>= 1) {
            if (threadIdx.x < s) red[threadIdx.x] += red[threadIdx.x + s];
            __syncthreads();
        }
        res[k] = red[0];
        __syncthreads();
    }
    if (threadIdx.x == 0) {
        const float BN = (float)((double)Brows * (double)N_DOS);
        const float Bf = (float)Brows;
        float total = res[0] / BN                     // dos_loss
                    + 0.05f  * (res[3] / Bf)          // scaling_loss
                    + 0.005f * (res[1] / BN)          // cumsum_loss
                    + 0.15f  * (res[2] / (Bf * 5.f)); // features_loss
        out_scalar[0] = total;
    }
}

extern "C" void kernel_launch(void* const* d_in, const int* in_sizes, int n_in,
                              void* d_out, int out_size, void* d_ws, size_t ws_size,
                              hipStream_t stream) {
    const float* outp     = (const float*)d_in[0];
    const float* scaling  = (const float*)d_in[1];
    const float* scaled   = (const float*)d_in[2];
    const float* sfac     = (const float*)d_in[3];
    const float* features = (const float*)d_in[4];

    const long long B = in_sizes[1];                       // 131072
    const int rows_per_block = WAVES_PER_BLOCK * ROWS_PER_WAVE;  // 128
    const int nblocks = (int)(B / rows_per_block);         // 1024

    float* partials = (float*)d_ws;                        // nblocks*4 floats

    dos_loss_main<<<nblocks, BLOCK_THREADS, 0, stream>>>(
        outp, scaling, scaled, sfac, features, partials);
    dos_loss_finish<<<1, 256, 0, stream>>>(partials, nblocks, B, (float*)d_out);
}